// PillarFeatureNet_66013647339880
// MI455X (gfx1250) — compile-verified
//
#include <hip/hip_runtime.h>

typedef __attribute__((ext_vector_type(2))) float v2f;
typedef __attribute__((ext_vector_type(8))) float v8f;

#define P_VOX   150000
#define N_PTS   20
#define C_IN    5
#define C_OUT   64
#define ROWS    (P_VOX * N_PTS)      /* 3,000,000 GEMM rows            */
#define TILE_ROWS 80                 /* 4 whole pillars per wave       */
#define NTILES  (ROWS / TILE_ROWS)   /* 37,500                         */

/* ws layout (floats):
   [0,64)   per-channel sum          (zeroed each call by k_means)
   [64,128) per-channel sum-of-sq
   [128,192) scale = gamma*rsqrt(var+eps)
   [192,256) shift = beta - mu*scale
   [256, 256+3P) per-pillar mean xyz                                   */

// ---------------------------------------------------------------------------
// Kernel 0: per-pillar mean of first 3 channels; also zero the stats slots.
// ---------------------------------------------------------------------------
__global__ void k_means(const float* __restrict__ vf,
                        const int* __restrict__ nums,
                        float* __restrict__ ws) {
  int gid = blockIdx.x * blockDim.x + threadIdx.x;
  if (gid < 128) ws[gid] = 0.0f;            // zero sum / sumsq
  if (gid >= P_VOX) return;
  const float* base = vf + (size_t)gid * (N_PTS * C_IN);
  float sx = 0.f, sy = 0.f, sz = 0.f;
#pragma unroll
  for (int j = 0; j < N_PTS; ++j) {
    sx += base[j * C_IN + 0];
    sy += base[j * C_IN + 1];
    sz += base[j * C_IN + 2];
  }
  float inv = 1.0f / (float)nums[gid];
  float* m = ws + 256 + (size_t)gid * 3;
  m[0] = sx * inv; m[1] = sy * inv; m[2] = sz * inv;
}

// ---------------------------------------------------------------------------
// Shared helpers for the WMMA GEMM (x9 row -> A operand, W -> B operand)
// ---------------------------------------------------------------------------
__device__ __forceinline__ void compute_A(const float* __restrict__ vf,
                                          const int* __restrict__ coords,
                                          const float* __restrict__ means,
                                          int row, bool laneLo,
                                          float A0[3], float A1[3]) {
  const int p = row / N_PTS;
  const float* f = vf + (size_t)row * C_IN;
  const float f0 = f[0], f1 = f[1], f2 = f[2];
  const float* m = means + (size_t)p * 3;
  const float ysv = (float)coords[p * 4 + 2];
  const float xsv = (float)coords[p * 4 + 3];
  const float cx = f0 - ((xsv + 0.5f) * 0.16f + 0.0f);
  const float cy = f1 - ((ysv + 0.5f) * 0.16f - 40.0f);
  const float cz = f2 + 1.0f;                 // f2 - (0.5*4 - 3)
  const float w0 = (f0 != 0.0f) ? 1.0f : 0.0f;
  const float w1 = (f1 != 0.0f) ? 1.0f : 0.0f;
  const float w2 = (f2 != 0.0f) ? 1.0f : 0.0f;
  const float x0 = f0, x1 = f1, x2 = f2;
  const float x3 = cx * w0, x4 = cy * w1, x5 = cz * w2;
  const float x6 = (f0 - m[0]) * w0, x7 = (f1 - m[1]) * w1, x8 = (f2 - m[2]) * w2;
  // K padded 9 -> 12; lanes 0-15 carry K=4s,4s+1 and lanes 16-31 K=4s+2,4s+3.
  A0[0] = laneLo ? x0 : x2;  A1[0] = laneLo ? x1 : x3;
  A0[1] = laneLo ? x4 : x6;  A1[1] = laneLo ? x5 : x7;
  A0[2] = laneLo ? x8 : 0.f; A1[2] = 0.f;
}

__device__ __forceinline__ void load_B(const float* __restrict__ W, int lane,
                                       float B0[4][3], float B1[4][3]) {
  const int col16 = lane & 15;
  const bool lo = lane < 16;
#pragma unroll
  for (int t = 0; t < 4; ++t) {
    const float* wr = W + (t * 16 + col16) * 9;  // B[k][col] = W[col*9+k]
    B0[t][0] = lo ? wr[0] : wr[2];
    B1[t][0] = lo ? wr[1] : wr[3];
    B0[t][1] = lo ? wr[4] : wr[6];
    B1[t][1] = lo ? wr[5] : wr[7];
    B0[t][2] = lo ? wr[8] : 0.0f;
    B1[t][2] = 0.0f;
  }
}

__device__ __forceinline__ v8f mma3(const float A0[3], const float A1[3],
                                    const float B0[3], const float B1[3]) {
  v8f c = {};
#pragma unroll
  for (int s = 0; s < 3; ++s) {
    v2f a; a.x = A0[s]; a.y = A1[s];
    v2f b; b.x = B0[s]; b.y = B1[s];
    c = __builtin_amdgcn_wmma_f32_16x16x4_f32(false, a, false, b,
                                              (short)0, c, false, false);
  }
  return c;
}

// ---------------------------------------------------------------------------
// Kernel 1: GEMM pass for BN statistics (sum, sum of squares per channel).
// ---------------------------------------------------------------------------
__global__ void __launch_bounds__(256) k_stats(const float* __restrict__ vf,
                                               const int* __restrict__ coords,
                                               const float* __restrict__ W,
                                               float* __restrict__ ws) {
  __shared__ float s_stats[128];
  const int tid = threadIdx.x;
  const int lane = tid & 31;
  const bool lo = lane < 16;
  if (tid < 128) s_stats[tid] = 0.0f;
  __syncthreads();

  const float* means = ws + 256;
  float B0[4][3], B1[4][3];
  load_B(W, lane, B0, B1);

  float sumAcc[4] = {0.f, 0.f, 0.f, 0.f};
  float sqAcc[4]  = {0.f, 0.f, 0.f, 0.f};

  const int gw = blockIdx.x * 8 + (tid >> 5);
  const int stride = gridDim.x * 8;
  for (int tile = gw; tile < NTILES; tile += stride) {   // wave-uniform loop
    const int rowBase = tile * TILE_ROWS;
#pragma unroll
    for (int q = 0; q < 5; ++q) {
      float A0[3], A1[3];
      compute_A(vf, coords, means, rowBase + 16 * q + (lane & 15), lo, A0, A1);
#pragma unroll
      for (int t = 0; t < 4; ++t) {
        v8f c = mma3(A0, A1, B0[t], B1[t]);
#pragma unroll
        for (int i = 0; i < 8; ++i) {      // 8 rows, all in channel t*16+(lane&15)
          float v = c[i];
          sumAcc[t] += v;
          sqAcc[t]  += v * v;
        }
      }
    }
  }
#pragma unroll
  for (int t = 0; t < 4; ++t) {            // fold lane pairs (l, l+16): same channel
    float s = sumAcc[t] + __shfl_xor(sumAcc[t], 16, 32);
    float q = sqAcc[t]  + __shfl_xor(sqAcc[t], 16, 32);
    if (lo) {
      atomicAdd(&s_stats[t * 16 + lane], s);
      atomicAdd(&s_stats[64 + t * 16 + lane], q);
    }
  }
  __syncthreads();
  if (tid < 128) atomicAdd(&ws[tid], s_stats[tid]);   // 128 global atomics / WG
}

// ---------------------------------------------------------------------------
// Kernel 2: fold stats + gamma/beta into per-channel scale & shift.
// ---------------------------------------------------------------------------
__global__ void k_finalize(const float* __restrict__ gamma,
                           const float* __restrict__ beta,
                           float* __restrict__ ws) {
  int c = threadIdx.x;
  if (c >= C_OUT) return;
  const float invN = 1.0f / (float)ROWS;
  float mu  = ws[c] * invN;
  float var = ws[64 + c] * invN - mu * mu;
  float rs  = rsqrtf(var + 1e-3f);
  float sc  = rs * gamma[c];
  ws[128 + c] = sc;
  ws[192 + c] = beta[c] - mu * sc;
}

// ---------------------------------------------------------------------------
// Kernel 3: GEMM pass 2 + BN + ReLU + per-pillar max + coords + grid_size.
// One wave = 80 rows = 4 whole pillars -> max never crosses a wave.
// ---------------------------------------------------------------------------
__global__ void __launch_bounds__(256) k_out(const float* __restrict__ vf,
                                             const int* __restrict__ coords,
                                             const float* __restrict__ W,
                                             const float* __restrict__ ws,
                                             float* __restrict__ out) {
  const int tid = threadIdx.x;
  const int lane = tid & 31;
  const int tile = blockIdx.x * 8 + (tid >> 5);
  if (blockIdx.x == 0 && tid == 0) {       // grid_size = [500, 440]
    out[(size_t)P_VOX * 64 + (size_t)P_VOX * 3 + 0] = 500.0f;
    out[(size_t)P_VOX * 64 + (size_t)P_VOX * 3 + 1] = 440.0f;
  }
  if (tile >= NTILES) return;              // wave-uniform guard

  const bool lo = lane < 16;
  const float* means = ws + 256;
  float B0[4][3], B1[4][3];
  load_B(W, lane, B0, B1);
  float myScale[4], myShift[4];
#pragma unroll
  for (int t = 0; t < 4; ++t) {
    myScale[t] = ws[128 + t * 16 + (lane & 15)];
    myShift[t] = ws[192 + t * 16 + (lane & 15)];
  }
  float pmax[4][4];                        // [col-tile][pillar-in-tile]; ReLU>=0 so 0 is neutral
#pragma unroll
  for (int t = 0; t < 4; ++t)
#pragma unroll
    for (int j = 0; j < 4; ++j) pmax[t][j] = 0.0f;

  const int rowBase = tile * TILE_ROWS;
#pragma unroll
  for (int q = 0; q < 5; ++q) {
    float A0[3], A1[3];
    compute_A(vf, coords, means, rowBase + 16 * q + (lane & 15), lo, A0, A1);
#pragma unroll
    for (int t = 0; t < 4; ++t) {
      v8f c = mma3(A0, A1, B0[t], B1[t]);
#pragma unroll
      for (int i = 0; i < 8; ++i) {
        float y = fmaxf(fmaf(c[i], myScale[t], myShift[t]), 0.0f);
        const int jLo = (16 * q + i) / 20;       // compile-time after unroll
        const int jHi = (16 * q + 8 + i) / 20;   // (lanes>=16 hold rows M=8+i)
        if (jLo == jHi) {
          pmax[t][jLo] = fmaxf(pmax[t][jLo], y);
        } else {
          pmax[t][jLo] = fmaxf(pmax[t][jLo], lo ? y : 0.0f);
          pmax[t][jHi] = fmaxf(pmax[t][jHi], lo ? 0.0f : y);
        }
      }
    }
  }

  const int p0 = tile * 4;
#pragma unroll
  for (int t = 0; t < 4; ++t)
#pragma unroll
    for (int j = 0; j < 4; ++j) {
      float v = pmax[t][j];
      v = fmaxf(v, __shfl_xor(v, 16, 32));       // merge row-halves
      if (lo) out[(size_t)(p0 + j) * 64 + t * 16 + lane] = v;
    }

  if (lane < 4) {                                // coords_out = [b, y, x] as float
    int p = p0 + lane;
    float* co = out + (size_t)P_VOX * 64 + (size_t)p * 3;
    co[0] = (float)coords[p * 4 + 0];
    co[1] = (float)coords[p * 4 + 2];
    co[2] = (float)coords[p * 4 + 3];
  }
}

// ---------------------------------------------------------------------------
extern "C" void kernel_launch(void* const* d_in, const int* in_sizes, int n_in,
                              void* d_out, int out_size, void* d_ws, size_t ws_size,
                              hipStream_t stream) {
  const float* vf     = (const float*)d_in[0];
  const int*   nums   = (const int*)d_in[1];
  const int*   coords = (const int*)d_in[2];
  const float* W      = (const float*)d_in[3];
  const float* gamma  = (const float*)d_in[4];
  const float* beta   = (const float*)d_in[5];
  float* out = (float*)d_out;
  float* ws  = (float*)d_ws;

  k_means<<<dim3((P_VOX + 255) / 256), dim3(256), 0, stream>>>(vf, nums, ws);
  k_stats<<<dim3(512), dim3(256), 0, stream>>>(vf, coords, W, ws);
  k_finalize<<<dim3(1), dim3(64), 0, stream>>>(gamma, beta, ws);
  k_out<<<dim3((NTILES + 7) / 8), dim3(256), 0, stream>>>(vf, coords, W, ws, out);
}